// Coords2RMSD_20693152432407
// MI455X (gfx1250) — compile-verified
//
#include <hip/hip_runtime.h>
#include <cstdint>

// ---------------------------------------------------------------------------
// Masked Kabsch RMSD, TWO waves (64 threads) per sample, alternating tiles.
//  - Streams coords via global_load_async_to_lds_b128 (double buffered per wave,
//    ASYNCcnt) -> 2 waves/sample keeps ~6KB/sample of HBM transfers in flight.
//  - Accumulates ALL sufficient statistics with one V_WMMA_F32_16X16X4_F32 chain
//    A rows 0-2 = x, 3-5 = y, 6 = mask;  B cols 0-2 = y, 3-5 = x, 6 = mask
//    => D holds Sum(x y^T), Gram matrices, Sx, Sy, n in one accumulator.
//  - Full tiles run with zero masking ops; only the boundary tile masks.
//  - Partial accumulators combined in LDS; analytic 3x3 eigen-solve on lane 0.
// ---------------------------------------------------------------------------

#define SAMPLE_F    3072                   // floats per sample per array (1024 atoms * 3)
#define TILE_ATOMS  128
#define TILE_BYTES  (TILE_ATOMS * 12)      // 1536 B per array per tile
#define TILE_F      (TILE_BYTES / 4)       // 384 floats
#define BUF_BYTES   (2 * TILE_BYTES)       // inp tile + tgt tile = 3072 B
#define BUF_F       (BUF_BYTES / 4)        // 768 floats
#define OUT_F       (4 * BUF_F)            // LDS float offset of the 2 x 7x8 results
#define LDS_F       (OUT_F + 128)

typedef float v2f __attribute__((ext_vector_type(2)));
typedef float v8f __attribute__((ext_vector_type(8)));

__device__ __forceinline__ void async_b128(uint32_t lds_addr, uint32_t voff, uint64_t sbase) {
  // copies 16 B from global SADDR + VADDR to LDS address VDST  (GVS form), tracked by ASYNCcnt
  asm volatile("global_load_async_to_lds_b128 %0, %1, %2"
               :: "v"(lds_addr), "v"(voff), "s"(sbase) : "memory");
}
__device__ __forceinline__ void wait_async0() {
  asm volatile("s_wait_asynccnt 0x0" ::: "memory");
}
__device__ __forceinline__ void wait_async6() {
  asm volatile("s_wait_asynccnt 0x6" ::: "memory");
}

// issue one tile: 3 x b128 per array per wave (32 lanes * 16B = 512B each)
__device__ __forceinline__ void issue_tile(uint32_t ldsBuf, uint64_t inpB, uint64_t tgtB,
                                           int t, int lane) {
  uint32_t lo   = (uint32_t)lane * 16u;
  uint32_t gOff = (uint32_t)t * (uint32_t)TILE_BYTES + lo;
#pragma unroll
  for (int i = 0; i < 3; ++i) {
    async_b128(ldsBuf + lo + i * 512u,              gOff + i * 512u, inpB);
    async_b128(ldsBuf + TILE_BYTES + lo + i * 512u, gOff + i * 512u, tgtB);
  }
}

// One tile of 128 atoms -> 32 chained WMMAs.  MASKED=false for interior tiles.
template <bool MASKED>
__device__ __forceinline__ v8f tile_mac(v8f acc, const float* __restrict__ buf,
                                        int tbase, int natoms, int hi,
                                        float wL, float wO, int baseA, int baseB) {
#pragma unroll 4
  for (int k = 0; k < TILE_ATOMS / 4; ++k) {
    const int a0 = (k << 2) + (hi << 1);     // local atom for K-slot 0 (and +1)
    const int i0 = 3 * a0;
    float rA0 = buf[i0 + baseA];
    float rA1 = buf[i0 + 3 + baseA];
    float rB0 = buf[i0 + baseB];
    float rB1 = buf[i0 + 3 + baseB];
    // branchless per-lane row/col shaping: data rows pass through, row 6 -> 1.0,
    // rows 7..15 -> 0.  (wL, wO are per-lane constants.)
    float A0 = fmaf(rA0, wL, wO);
    float A1 = fmaf(rA1, wL, wO);
    float B0 = fmaf(rB0, wL, wO);
    float B1 = fmaf(rB1, wL, wO);
    if (MASKED) {
      const int  g0 = tbase + a0;
      const bool v0 = (g0 < natoms);
      const bool v1 = (g0 + 1 < natoms);
      A0 = v0 ? A0 : 0.0f;  B0 = v0 ? B0 : 0.0f;
      A1 = v1 ? A1 : 0.0f;  B1 = v1 ? B1 : 0.0f;
    }
    v2f Aop; Aop.x = A0; Aop.y = A1;
    v2f Bop; Bop.x = B0; Bop.y = B1;
    acc = __builtin_amdgcn_wmma_f32_16x16x4_f32(false, Aop, false, Bop,
                                                (short)0, acc, false, false);
  }
  return acc;
}

__global__ void __launch_bounds__(64)
kabsch_rmsd_wmma(const float* __restrict__ inp, const float* __restrict__ tgt,
                 const int* __restrict__ num_atoms, float* __restrict__ out) {
  __shared__ __align__(16) float lds[LDS_F];

  const int b      = blockIdx.x;
  const int tid    = threadIdx.x;
  const int lane   = tid & 31;
  const int w      = tid >> 5;                     // wave id (0/1)
  const int natoms = num_atoms[b];
  const int tiles  = (natoms + TILE_ATOMS - 1) / TILE_ATOMS;
  const int fullT  = natoms / TILE_ATOMS;          // tiles with no masking needed

  const uint64_t inpB = (uint64_t)(uintptr_t)(inp + (size_t)b * SAMPLE_F);
  const uint64_t tgtB = (uint64_t)(uintptr_t)(tgt + (size_t)b * SAMPLE_F);
  // each wave owns two buffers: wave w region at [w*2*BUF_BYTES, +2*BUF_BYTES)
  const uint32_t ldsU = (uint32_t)(uintptr_t)(&lds[0]) + (uint32_t)(w * 2 * BUF_BYTES);
  const float*   wbuf = &lds[w * 2 * BUF_F];

  // ---- per-lane constant operand selectors -------------------------------
  const int   m  = lane & 15;          // A row / B col
  const int   hi = lane >> 4;          // K half (0 -> k0,k1 ; 1 -> k2,k3)
  const float wLc = (m < 6) ? 1.0f : 0.0f;   // pass loaded value
  const float wOc = (m == 6) ? 1.0f : 0.0f;  // mask row/col constant 1
  // A rows 0-2: x comp m (inp @0) ; rows 3-5: y comp m-3 (tgt @TILE_F)
  const int baseA = (m < 3) ? m : ((m < 6) ? (TILE_F + (m - 3)) : 0);
  // B cols 0-2: y comp m (tgt)   ; cols 3-5: x comp m-3 (inp)
  const int baseB = (m < 3) ? (TILE_F + m) : ((m < 6) ? (m - 3) : 0);

  v8f acc = {0.f, 0.f, 0.f, 0.f, 0.f, 0.f, 0.f, 0.f};

  if (w < tiles) issue_tile(ldsU, inpB, tgtB, w, lane);

  // wave w processes tiles w, w+2, w+4, ... with its private double buffer
  for (int t = w; t < tiles; t += 2) {
    const int bi = ((t - w) >> 1) & 1;
    if (t + 2 < tiles) {
      issue_tile(ldsU + (uint32_t)((bi ^ 1) * BUF_BYTES), inpB, tgtB, t + 2, lane);
      wait_async6();                   // current tile's 6 transfers retired
    } else {
      wait_async0();
    }
    const float* bufp = wbuf + bi * BUF_F;
    if (t < fullT) {
      acc = tile_mac<false>(acc, bufp, 0, natoms, hi, wLc, wOc, baseA, baseB);
    } else {
      acc = tile_mac<true>(acc, bufp, t * TILE_ATOMS, natoms, hi, wLc, wOc, baseA, baseB);
    }
  }

  // ---- dump D[0..7][0..6] per wave (lane n holds column n in acc[0..7]) --
  if (lane < 7) {
#pragma unroll
    for (int r = 0; r < 8; ++r) lds[OUT_F + w * 64 + lane * 8 + r] = acc[r];
  }
  __syncthreads();

  if (tid == 0) {
#define Dm(r, c) (lds[OUT_F + (c) * 8 + (r)] + lds[OUT_F + 64 + (c) * 8 + (r)])
    const float nf    = (float)natoms;
    const float inv_n = 1.0f / nf;
    float Sx0 = Dm(0, 6), Sx1 = Dm(1, 6), Sx2 = Dm(2, 6);
    float Sy0 = Dm(6, 0), Sy1 = Dm(6, 1), Sy2 = Dm(6, 2);
    // R = Sum(x y^T) - Sx Sy^T / n
    float R00 = Dm(0,0) - Sx0*Sy0*inv_n, R01 = Dm(0,1) - Sx0*Sy1*inv_n, R02 = Dm(0,2) - Sx0*Sy2*inv_n;
    float R10 = Dm(1,0) - Sx1*Sy0*inv_n, R11 = Dm(1,1) - Sx1*Sy1*inv_n, R12 = Dm(1,2) - Sx1*Sy2*inv_n;
    float R20 = Dm(2,0) - Sx2*Sy0*inv_n, R21 = Dm(2,1) - Sx2*Sy1*inv_n, R22 = Dm(2,2) - Sx2*Sy2*inv_n;
    float exx = (Dm(0,3) + Dm(1,4) + Dm(2,5)) - (Sx0*Sx0 + Sx1*Sx1 + Sx2*Sx2) * inv_n;
    float eyy = (Dm(3,0) + Dm(4,1) + Dm(5,2)) - (Sy0*Sy0 + Sy1*Sy1 + Sy2*Sy2) * inv_n;
    float e   = exx + eyy;
    float det = R00*(R11*R22 - R12*R21) - R01*(R10*R22 - R12*R20) + R02*(R10*R21 - R11*R20);
    // A = R^T R (symmetric), singular values = sqrt(eig(A))
    float A00 = R00*R00 + R10*R10 + R20*R20;
    float A11 = R01*R01 + R11*R11 + R21*R21;
    float A22 = R02*R02 + R12*R12 + R22*R22;
    float A01 = R00*R01 + R10*R11 + R20*R21;
    float A02 = R00*R02 + R10*R12 + R20*R22;
    float A12 = R01*R02 + R11*R12 + R21*R22;
    float q  = (A00 + A11 + A22) * (1.0f / 3.0f);
    float p1 = A01*A01 + A02*A02 + A12*A12;
    float c0 = A00 - q, c1 = A11 - q, c2 = A22 - q;
    float p2 = c0*c0 + c1*c1 + c2*c2 + 2.0f * p1;
    float p  = sqrtf(fmaxf(p2, 0.0f) * (1.0f / 6.0f));
    float ip = (p > 1e-20f) ? (1.0f / p) : 0.0f;
    float b00 = c0*ip, b11 = c1*ip, b22 = c2*ip;
    float b01 = A01*ip, b02 = A02*ip, b12 = A12*ip;
    float detB = b00*(b11*b22 - b12*b12) - b01*(b01*b22 - b12*b02) + b02*(b01*b12 - b11*b02);
    float r = fminf(fmaxf(0.5f * detB, -1.0f), 1.0f);
    float phi = acosf(r) * (1.0f / 3.0f);
    float l1 = q + 2.0f * p * cosf(phi);
    float l3 = q + 2.0f * p * cosf(phi + 2.0943951023931953f);  // +2*pi/3 (smallest)
    float l2 = 3.0f * q - l1 - l3;
    float s1 = sqrtf(fmaxf(l1, 0.0f));
    float s2 = sqrtf(fmaxf(l2, 0.0f));
    float s3 = sqrtf(fmaxf(l3, 0.0f));
    float dsign = (det >= 0.0f) ? 1.0f : -1.0f;
    float s = s1 + s2 + dsign * s3;
    float msd = (e - 2.0f * s) * inv_n;
    out[b] = sqrtf(fmaxf(msd, 1e-12f));
#undef Dm
  }
}

extern "C" void kernel_launch(void* const* d_in, const int* in_sizes, int n_in,
                              void* d_out, int out_size, void* d_ws, size_t ws_size,
                              hipStream_t stream) {
  const float* inp    = (const float*)d_in[0];
  const float* tgt    = (const float*)d_in[1];
  const int*   natoms = (const int*)d_in[2];
  float*       out    = (float*)d_out;
  const int batch = in_sizes[2];   // num_atoms has one entry per sample
  kabsch_rmsd_wmma<<<batch, 64, 0, stream>>>(inp, tgt, natoms, out);
}